// AttentionCTCLoss_86294482911706
// MI455X (gfx1250) — compile-verified
//
#include <hip/hip_runtime.h>
#include <math.h>

#define NEG_INF  (-1e30f)
#define KDIM     512
#define SMAXPAD  1056   // 1025 states padded

// ---------------- wave32 reductions (no barriers) ----------------
__device__ __forceinline__ float wave_max(float v) {
#pragma unroll
    for (int off = 16; off >= 1; off >>= 1)
        v = fmaxf(v, __shfl_xor(v, off, 32));
    return v;
}
__device__ __forceinline__ float wave_sum(float v) {
#pragma unroll
    for (int off = 16; off >= 1; off >>= 1)
        v += __shfl_xor(v, off, 32);
    return v;
}

// ---------------- Kernel A: logZ[b,t], one wave per row ----------------
// logits row = [-1.0 (blank), attn[b,0,t,0..in_len-1]], keys > in_len masked.
// Masked entries contribute exp(-1e9 - m) == 0.0f exactly (float underflow),
// so masking reduces to a length check. float4 (b128) loads: 4 VMEM per lane.
// 2D grid (x: rows-of-8, y: batch) -> no integer division.
__global__ __launch_bounds__(256)
void ctc_logz_kernel(const float* __restrict__ attn,
                     const int*   __restrict__ in_lens,
                     float*       __restrict__ logZ,
                     int T) {
    const int wave = threadIdx.x >> 5;
    const int lane = threadIdx.x & 31;
    const int b    = blockIdx.y;
    const int t    = blockIdx.x * 8 + wave;
    if (t >= T) return;
    const int in_len = in_lens[b];
    const float4* rp = (const float4*)(attn + ((size_t)b * T + t) * KDIM);

    float x[16];
    float m = -1.0f;  // blank logit always present
#pragma unroll
    for (int i = 0; i < 4; ++i) {
        const float4 v4  = rp[lane + i * 32];          // 512B contiguous per load
        const int   base = (lane + i * 32) * 4;        // element index in row
        x[4*i + 0] = (base + 0 < in_len) ? v4.x : NEG_INF;
        x[4*i + 1] = (base + 1 < in_len) ? v4.y : NEG_INF;
        x[4*i + 2] = (base + 2 < in_len) ? v4.z : NEG_INF;
        x[4*i + 3] = (base + 3 < in_len) ? v4.w : NEG_INF;
        m = fmaxf(m, fmaxf(fmaxf(x[4*i], x[4*i+1]), fmaxf(x[4*i+2], x[4*i+3])));
    }
    m = wave_max(m);
    float s = 0.0f;
#pragma unroll
    for (int i = 0; i < 16; ++i) s += __expf(x[i] - m);  // exp(-1e30 - m) == 0
    s = wave_sum(s);
    s += __expf(-1.0f - m);                              // blank term, added once
    if (lane == 0) logZ[(size_t)b * T + t] = m + __logf(s);
}

// ---------------- Kernel B: serial CTC scan, one workgroup per batch ----------------
// Thread tid owns the state PAIR (s0 = 2*tid even, s1 = 2*tid+1 odd):
//  - uniform control flow: even state = 2-term logaddexp, odd = 3-term
//  - A[2t],A[2t+1] read/written as aligned float2 (ds b64 ops)
//  - odd emit read row[tid]: fully coalesced contiguous global load
// alpha double-buffered in LDS -> exactly ONE barrier per timestep.
// Invalid states (s > 2*in_len) stay NEG_INF forever; validity is a prefix and
// the recurrence only reads s-1, s-2, so they never feed valid states.
__global__ __launch_bounds__(512)
void ctc_scan_kernel(const float* __restrict__ attn,
                     const int*   __restrict__ in_lens,
                     const int*   __restrict__ out_lens,
                     const float* __restrict__ logZ,
                     float*       __restrict__ loss_out,
                     int T) {
    __shared__ float A[2][SMAXPAD];

    const int b      = blockIdx.x;
    const int tid    = threadIdx.x;            // 0..511
    const int in_len = in_lens[b];
    int out_len      = out_lens[b];
    if (out_len > T) out_len = T;
    const int Smax   = 2 * in_len;             // last valid state index
    const float* ap  = attn + (size_t)b * T * KDIM;
    const float* lz  = logZ + (size_t)b * T;

    // init both buffers to NEG_INF, then t=0 states
    A[0][tid]       = NEG_INF;  A[1][tid]       = NEG_INF;
    A[0][tid + 512] = NEG_INF;  A[1][tid + 512] = NEG_INF;
    if (tid < SMAXPAD - 1024) { A[0][1024 + tid] = NEG_INF; A[1][1024 + tid] = NEG_INF; }
    __syncthreads();
    if (tid == 0) {
        const float l0 = lz[0];
        A[0][0] = -1.0f - l0;     // blank emit at t=0
        A[0][1] = ap[0] - l0;     // first label emit at t=0 (always valid)
    }
    __syncthreads();

    const bool has1024 = (1024 <= Smax);       // in_len == 512
    int p = 0;
    for (int t = 1; t < out_len; ++t) {
        const float* row = ap + (size_t)t * KDIM;
        // gfx1250 global_prefetch_b8: stream next timestep's 2KB row (32 x 64B lines)
        if (tid < 32 && (t + 1) < out_len)
            __builtin_prefetch(row + KDIM + tid * 16, 0, 3);

        const float l = lz[t];

        const float2 a01 = *(const float2*)&A[p][2 * tid];          // A[2t], A[2t+1]
        const float  am1 = (tid >= 1) ? A[p][2 * tid - 1] : NEG_INF; // A[2t-1]

        // even state s0 = 2*tid: transitions from s0, s0-1 (no skip)
        const float M0 = fmaxf(a01.x, am1);
        const float v0 = M0 + __logf(__expf(a01.x - M0) + __expf(am1 - M0));
        const float n0 = v0 + (-1.0f - l);                           // blank emit

        // odd state s1 = 2*tid+1: transitions from s1, s1-1, s1-2 (skip ok iff tid>=1,
        // and am1 is already NEG_INF for tid==0)
        const float M1 = fmaxf(a01.y, fmaxf(a01.x, am1));
        const float v1 = M1 + __logf(__expf(a01.y - M1) + __expf(a01.x - M1) + __expf(am1 - M1));
        const float n1 = v1 + (row[tid] - l);                        // label emit, coalesced

        if (tid < in_len) {
            float2 nw; nw.x = n0; nw.y = n1;
            *(float2*)&A[p ^ 1][2 * tid] = nw;                       // ds_store_b64
        } else if (tid == in_len) {
            A[p ^ 1][2 * tid] = n0;                                  // s0 == Smax only
        }
        if (tid == 0 && has1024) {                                   // state 1024 (even)
            const float a0 = A[p][1024];
            const float a1 = A[p][1023];
            const float M  = fmaxf(a0, a1);
            A[p ^ 1][1024] = M + __logf(__expf(a0 - M) + __expf(a1 - M)) + (-1.0f - l);
        }
        __syncthreads();     // single barrier: all reads of A[p] happened above
        p ^= 1;
    }

    if (tid == 0) {
        const float e1 = A[p][Smax];
        const float e2 = A[p][Smax - 1];
        const float M  = fmaxf(e1, e2);
        float L = -(M + __logf(__expf(e1 - M) + __expf(e2 - M)));
        if (isnan(L) || L > 1e29f) L = 0.0f;
        loss_out[b] = L / (float)in_len;
    }
}

// ---------------- Kernel C: deterministic mean over B losses (one wave) ----------------
__global__ void ctc_mean_kernel(const float* __restrict__ loss,
                                float* __restrict__ out, int B) {
    const int lane = threadIdx.x;   // 32 threads
    float v = 0.0f;
    for (int i = lane; i < B; i += 32) v += loss[i];
    v = wave_sum(v);
    if (lane == 0) out[0] = v / (float)B;
}

extern "C" void kernel_launch(void* const* d_in, const int* in_sizes, int n_in,
                              void* d_out, int out_size, void* d_ws, size_t ws_size,
                              hipStream_t stream) {
    const float* attn     = (const float*)d_in[0];
    const int*   in_lens  = (const int*)d_in[1];
    const int*   out_lens = (const int*)d_in[2];

    const int B = in_sizes[1];                       // batch = len(in_lens)
    const int T = in_sizes[0] / (B * KDIM);          // attn is (B,1,T,K), K=512

    float* logZ   = (float*)d_ws;                    // B*T floats
    float* loss_b = logZ + (size_t)B * T;            // B floats

    // A: one wave32 per row, 8 rows per 256-thread block; grid.y = batch
    dim3 gridA((T + 7) / 8, B);
    ctc_logz_kernel<<<gridA, 256, 0, stream>>>(attn, in_lens, logZ, T);

    // B: one 512-thread workgroup per batch element (serial scan, state pairs)
    ctc_scan_kernel<<<B, 512, 0, stream>>>(attn, in_lens, out_lens, logZ, loss_b, T);

    // C: single wave mean
    ctc_mean_kernel<<<1, 32, 0, stream>>>(loss_b, (float*)d_out, B);

    (void)n_in; (void)out_size; (void)ws_size;
}